// knnModule_16733192585417
// MI455X (gfx1250) — compile-verified
//
#include <hip/hip_runtime.h>
#include <hip/hip_bf16.h>
#include <math.h>

typedef __attribute__((ext_vector_type(2))) float        v2f;
typedef __attribute__((ext_vector_type(8))) float        v8f;
typedef __attribute__((ext_vector_type(4))) unsigned int v4u;
typedef __attribute__((ext_vector_type(8))) int          v8i;
typedef __attribute__((ext_vector_type(4))) int          v4i;

#define B_   4
#define N_   8192
#define M_   8192
#define D_   128
#define K_   16
#define TILE_M 64
#define ROWS_PER_BLOCK 128
#define YT_STRIDE 132   /* 128 + 4 floats pad -> conflict-free ds_load_b64 B reads */

#define HAVE_TDM __has_builtin(__builtin_amdgcn_tensor_load_to_lds)

#if HAVE_TDM
// Issue one TDM DMA: 64 rows x 128 f32 from global into LDS with a 4-DWORD pad
// inserted every 128 DWORDs (hardware synthesizes the 132-float padded stride).
__device__ __forceinline__ void tdm_issue_tile(const float* gsrc, unsigned ldsOff) {
    unsigned long long ga = (unsigned long long)(size_t)gsrc;
    v4u g0;
    g0[0] = 1u;                                       // count=1, user mode, no gather
    g0[1] = ldsOff;                                   // lds_addr (bytes)
    g0[2] = (unsigned)(ga & 0xFFFFFFFFu);             // global_addr[31:0]
    g0[3] = (unsigned)((ga >> 32) & 0x01FFFFFFu)      // global_addr[56:32]
          | (2u << 30);                               // type=2 ("image")
    v8i g1;
    g1[0] = (2 << 16)                                 // data_size = 4 bytes
          | (1 << 20)                                 // pad_enable
          | (6 << 22)                                 // pad_interval: every 128 DWORDs
          | (3 << 25);                                // pad_amount: 4 DWORDs
    g1[1] = (int)(128u << 16);                        // tensor_dim0[15:0] in [31:16]
    g1[2] = (int)(64u << 16);                         // tensor_dim0[31:16]=0 | tensor_dim1[15:0]=64
    g1[3] = (int)(128u << 16);                        // tensor_dim1[31:16]=0 | tile_dim0=128
    g1[4] = 64;                                       // tile_dim1=64, tile_dim2=0
    g1[5] = 128;                                      // tensor_dim0_stride[31:0]=128
    g1[6] = 0;
    g1[7] = 0;                                        // dim1 stride unused (2D)
    v4i z4 = {0, 0, 0, 0};
#if defined(__clang_major__) && (__clang_major__ >= 23)
    v8i z8 = {0, 0, 0, 0, 0, 0, 0, 0};
    __builtin_amdgcn_tensor_load_to_lds(g0, g1, z4, z4, z8, 0);
#else
    __builtin_amdgcn_tensor_load_to_lds(g0, g1, z4, z4, 0);
#endif
}
#endif

__global__ __launch_bounds__(256) void knn_topk_wmma_f32(
    const float* __restrict__ xg, const float* __restrict__ yg,
    float* __restrict__ odist, int* __restrict__ oidx)
{
    __shared__ float yt[2][TILE_M * YT_STRIDE];   // 2 x 33792 B double buffer; buf0 reused for merge
    __shared__ float y2s[2][TILE_M];
    __shared__ float ktile[8][16 * 17];           // per-wave transpose tile (padded)
    __shared__ float x2part[8][32];

    const int tid  = threadIdx.x;
    const int wave = tid >> 5;
    const int lane = tid & 31;
    const int lrow = lane & 15;   // 0..15
    const int hi   = lane >> 4;   // 0 or 1

    const int blk     = blockIdx.x;                 // B * (N/128) blocks
    const int bb      = blk / (N_ / ROWS_PER_BLOCK);
    const int nTile   = blk % (N_ / ROWS_PER_BLOCK);
    const int rowBase = nTile * ROWS_PER_BLOCK + wave * 16;
    const int row     = rowBase + lrow;

    // ---- A fragments: x rows resident in registers for the whole M sweep ----
    // 32-bit A 16x4 layout: lanes 0-15 hold K=k0,k0+1 ; lanes 16-31 hold K=k0+2,k0+3
    v2f a[32];
    float x2p = 0.f;
    {
        const float* xrow = xg + ((size_t)bb * N_ + row) * D_;
        #pragma unroll
        for (int i = 0; i < 32; ++i) {
            a[i] = *(const v2f*)(xrow + 4 * i + 2 * hi);
            x2p += a[i].x * a[i].x + a[i].y * a[i].y;
        }
    }
    x2part[wave][lane] = x2p;   // halves summed at output time

    // ---- per-lane running top-K over this lane's column half ----
    float bd[K_]; int bi[K_];
    #pragma unroll
    for (int j = 0; j < K_; ++j) { bd[j] = 3.0e38f; bi[j] = 0; }

    const float* ybatch = yg + (size_t)bb * M_ * D_;

#if HAVE_TDM
    // prime the pipeline: wave0 DMAs tile 0 into buffer 0
    if (wave == 0) tdm_issue_tile(ybatch, (unsigned)(size_t)&yt[0][0]);
#endif

    for (int it = 0; it < M_ / TILE_M; ++it) {
        const int mBase = it * TILE_M;
        const int buf   = it & 1;

#if HAVE_TDM
        if (wave == 0) __builtin_amdgcn_s_wait_tensorcnt(0);  // tile `it` resident
        __syncthreads();                                      // publish LDS tile to all waves
        if (mBase + TILE_M < M_)                              // overlap: DMA next tile
            if (wave == 0)
                tdm_issue_tile(ybatch + (size_t)(mBase + TILE_M) * D_,
                               (unsigned)(size_t)&yt[buf ^ 1][0]);
#else
        // fallback: manual staging [64 cols][128] into padded LDS (8 float4/thread)
        #pragma unroll
        for (int q = 0; q < (TILE_M * D_ / 4) / 256; ++q) {
            int e  = q * 256 + tid;
            int c  = e >> 5;
            int kq = e & 31;
            const float4 v = *(const float4*)(ybatch + (size_t)(mBase + c) * D_ + kq * 4);
            float* dst = &yt[buf][c * YT_STRIDE + kq * 4];
            dst[0] = v.x; dst[1] = v.y; dst[2] = v.z; dst[3] = v.w;
        }
        if (mBase + TILE_M < M_)
            __builtin_prefetch(ybatch + (size_t)(mBase + TILE_M + (tid >> 2)) * D_ + (tid & 3) * 32, 0, 0);
        __syncthreads();
#endif

        // y^2 per staged column (selection key constant)
        if (tid < TILE_M) {
            float s = 0.f;
            const float* r = &yt[buf][tid * YT_STRIDE];
            #pragma unroll 8
            for (int kk = 0; kk < D_; ++kk) s += r[kk] * r[kk];
            y2s[buf][tid] = s;
        }
        __syncthreads();

        #pragma unroll
        for (int sub = 0; sub < TILE_M / 16; ++sub) {
            // B fragment base: lane = column, K-parity by lane half (4x16 f32 B layout)
            const float* bbase = &yt[buf][(sub * 16 + lrow) * YT_STRIDE + 2 * hi];
            v8f c = {0.f, 0.f, 0.f, 0.f, 0.f, 0.f, 0.f, 0.f};
            #pragma unroll
            for (int i = 0; i < 32; ++i) {
                v2f bv = *(const v2f*)(bbase + 4 * i);
                c = __builtin_amdgcn_wmma_f32_16x16x4_f32(
                        /*neg_a=*/false, a[i], /*neg_b=*/false, bv,
                        /*c_mod=*/(short)0, c, /*reuse_a=*/false, /*reuse_b=*/false);
            }

            // D layout: lane holds col=lrow, rows r + 8*hi -> transpose via padded LDS
            float* kt = ktile[wave];
            #pragma unroll
            for (int r = 0; r < 8; ++r) kt[(hi * 8 + r) * 17 + lrow] = c[r];

            // lane owns row=lrow, cols hi*8 .. hi*8+7 ; key = y^2 - 2*x.y
            #pragma unroll
            for (int j = 0; j < 8; ++j) {
                int   ct   = hi * 8 + j;
                float xy   = kt[lrow * 17 + ct];
                float cand = y2s[buf][sub * 16 + ct] - 2.0f * xy;
                int   col  = mBase + sub * 16 + ct;
                if (cand < bd[K_ - 1]) {            // rarely taken after warm-up
                    float d = cand; int id = col;
                    #pragma unroll
                    for (int t = 0; t < K_; ++t) {  // sorted bubble-insert, registers only
                        if (d < bd[t]) {
                            float td = bd[t]; bd[t] = d;  d  = td;
                            int   ti = bi[t]; bi[t] = id; id = ti;
                        }
                    }
                }
            }
        }
#if !HAVE_TDM
        __syncthreads();   // before overwriting LDS with next tile
#endif
    }

    // ---- merge the two column-half lists per row, finalize, store ----
    __syncthreads();                          // everyone done with yt
    float* md = &yt[0][0] + wave * 1056;      // 8 waves * 1056 floats == 8448 floats of buf0
    int*   mi = (int*)(md + 512);
    #pragma unroll
    for (int j = 0; j < K_; ++j) { md[lane * K_ + j] = bd[j]; mi[lane * K_ + j] = bi[j]; }

    if (lane < 16) {
        const float* Ad = md + lane * K_;
        const float* Bd = md + (lane + 16) * K_;
        const int*   Ai = mi + lane * K_;
        const int*   Bi = mi + (lane + 16) * K_;
        float x2v = x2part[wave][lane] + x2part[wave][lane + 16];
        size_t obase = ((size_t)bb * N_ + (size_t)(rowBase + lane)) * K_;
        int ia = 0, ib = 0;
        for (int j = 0; j < K_; ++j) {
            float da = Ad[ia], db = Bd[ib];
            bool  ta = (da <= db);
            float d  = ta ? da : db;
            int   id = ta ? Ai[ia] : Bi[ib];
            if (ta) ++ia; else ++ib;
            float full = d + x2v;
            full = full < 0.f ? 0.f : full;
            odist[obase + j] = sqrtf(full);
            oidx [obase + j] = id;
        }
    }
}

extern "C" void kernel_launch(void* const* d_in, const int* in_sizes, int n_in,
                              void* d_out, int out_size, void* d_ws, size_t ws_size,
                              hipStream_t stream) {
    (void)in_sizes; (void)n_in; (void)out_size; (void)d_ws; (void)ws_size;
    const float* x = (const float*)d_in[0];   // [B,N,D] f32
    const float* y = (const float*)d_in[1];   // [B,M,D] f32
    float* odist = (float*)d_out;                                  // [B,N,K] f32
    int*   oidx  = (int*)((float*)d_out + (size_t)B_ * N_ * K_);   // [B,N,K] i32 (concat)

    dim3 grid(B_ * (N_ / ROWS_PER_BLOCK));   // 4 * 64 = 256 blocks
    dim3 block(256);                          // 8 wave32 per block
    knn_topk_wmma_f32<<<grid, block, 0, stream>>>(x, y, odist, oidx);
}